// RNNBase_16595753632519
// MI455X (gfx1250) — compile-verified
//
#include <hip/hip_runtime.h>
#include <math.h>

// ---------------------------------------------------------------------------
// Hyperbolic (Mobius) GRU, c = 1.  T=1024, B=64, I=H=256.
//   Stage 1: row norms of x   ->  f[row]=tanh(xn)/xn (expmap0 scale), un[row]
//   Stage 2: UX = mobius_matvec(W_i*, expmap0(x)) for all (t,b): WMMA f32 GEMM
//            [65536 x 256] x [256 x 768], expmap fused into A-load, Mobius
//            norm-scaling fused into the store (per-gate row norms in-block).
//   Stage 3: sequential scan, 4 workgroups x 16 batch rows (recurrence is
//            independent per batch row). h lives in LDS, two WMMA phases per
//            step, wave-per-row shuffle reductions for all Mobius norms.
// ---------------------------------------------------------------------------

typedef float v2f __attribute__((ext_vector_type(2)));
typedef float v8f __attribute__((ext_vector_type(8)));

constexpr int   Tn  = 1024;
constexpr int   Bn  = 64;
constexpr int   In  = 256;
constexpr int   Hn  = 256;
constexpr int   NR  = Tn * Bn;       // 65536 rows
constexpr int   G3  = 3 * Hn;        // 768
constexpr float EPSF   = 1e-15f;
constexpr float CLAMPF = 0.9999999f; // fp32 rendering of 1 - 1e-7

__device__ __forceinline__ float artanh_c(float z) {      // z >= 0 here
  z = fminf(z, CLAMPF);
  return 0.5f * logf((1.f + z) / (1.f - z));
}
__device__ __forceinline__ float wave_allred(float s) {
  #pragma unroll
  for (int off = 1; off < 32; off <<= 1) s += __shfl_xor(s, off, 32);
  return s;
}
__device__ __forceinline__ float dot8(const float* a, const float* b) {
  float s = 0.f;
  #pragma unroll
  for (int i = 0; i < 8; ++i) s += a[i] * b[i];
  return wave_allred(s);
}
// Mobius addition (c=1), one wave owns the whole 256-row (8 elems/lane)
__device__ __forceinline__ void mob_add(const float* x, const float* y, float* o) {
  float x2 = dot8(x, x), y2 = dot8(y, y), xy = dot8(x, y);
  float den = fmaxf(1.f + 2.f * xy + x2 * y2, EPSF);
  float cx = 1.f + 2.f * xy + y2, cy = 1.f - x2;
  #pragma unroll
  for (int i = 0; i < 8; ++i) o[i] = (cx * x[i] + cy * y[i]) / den;
}
// Mobius pointwise mul: tanh(||w*x||/xn * artanh(xn)) * (w*x)/||w*x||
__device__ __forceinline__ void mob_pmul(const float* w, const float* x, float xn, float* o) {
  float wx[8];
  #pragma unroll
  for (int i = 0; i < 8; ++i) wx[i] = w[i] * x[i];
  float wxn = sqrtf(fmaxf(dot8(wx, wx), EPSF));
  float fct = tanhf(wxn / xn * artanh_c(xn)) / wxn;
  #pragma unroll
  for (int i = 0; i < 8; ++i) o[i] = wx[i] * fct;
}

// One 16x16 output tile, K=256, fp32 WMMA (16x16x4). A in LDS (lda floats),
// B = rows [wrow0, wrow0+16) of a [*,256] row-major weight matrix (B = W^T).
__device__ __forceinline__ v8f wmma_16x256(const float* As, int lda,
                                           const float* wmat, int wrow0, int lane) {
  const int m  = lane & 15;
  const int kk = (lane >> 4) << 1;            // lanes 16-31 hold K=2,3
  const float* ar = As + m * lda + kk;
  const float* br = wmat + (size_t)(wrow0 + m) * In + kk;
  v8f d = {0.f, 0.f, 0.f, 0.f, 0.f, 0.f, 0.f, 0.f};
  #pragma unroll 4
  for (int k0 = 0; k0 < In; k0 += 4) {
    v2f a, b;
    a.x = ar[k0]; a.y = ar[k0 + 1];
    b.x = br[k0]; b.y = br[k0 + 1];
    d = __builtin_amdgcn_wmma_f32_16x16x4_f32(false, a, false, b, (short)0, d, false, false);
  }
  return d;
}

// ---------------- Stage 1: per-row norms of x ----------------
__global__ void k_rownorm(const float* __restrict__ x,
                          float* __restrict__ f, float* __restrict__ un) {
  const int lane = threadIdx.x & 31;
  const int row  = blockIdx.x * 8 + (threadIdx.x >> 5);
  const float* xr = x + (size_t)row * In;
  float ss = 0.f;
  #pragma unroll
  for (int i = 0; i < 8; ++i) { float v = xr[lane + 32 * i]; ss += v * v; }
  ss = wave_allred(ss);
  if (lane == 0) {
    float xn = sqrtf(fmaxf(ss, EPSF));
    float th = tanhf(xn);
    f[row]  = th / xn;                           // expmap0 row scale
    un[row] = sqrtf(fmaxf(th * th, EPSF));       // safe_norm(expmap0(x))
  }
}

// ---------------- Stage 2: UX GEMM + Mobius scaling ----------------
// grid = (NR/16, 3 gates), block = 512 (16 waves; wave w -> col tile w)
__global__ __launch_bounds__(512) void k_gemm_ih(
    const float* __restrict__ x, const float* __restrict__ wih,
    const float* __restrict__ f, const float* __restrict__ un,
    float* __restrict__ ux) {
  constexpr int LDA = In + 4;
  __shared__ float As[16 * LDA];
  __shared__ float rowss[16], facS[16], fS[16], unS[16];

  const int rowbase = blockIdx.x * 16;
  const int gate    = blockIdx.y;            // 0=r, 1=h, 2=z (weight row blocks)
  const int tid  = threadIdx.x;
  const int wave = tid >> 5, lane = tid & 31;

  if (tid < 16) {
    rowss[tid] = 0.f;
    fS[tid]  = f[rowbase + tid];
    unS[tid] = un[rowbase + tid];
  }
  __syncthreads();
  for (int idx = tid; idx < 16 * In; idx += 512) {
    int r = idx >> 8, k = idx & (In - 1);
    As[r * LDA + k] = x[(size_t)(rowbase + r) * In + k] * fS[r];
  }
  __syncthreads();

  v8f d = wmma_16x256(As, LDA, wih, gate * Hn + wave * 16, lane);

  // per-row sum of squares across the whole gate (block-wide)
  float s[8];
  #pragma unroll
  for (int r = 0; r < 8; ++r) s[r] = d[r] * d[r];
  #pragma unroll
  for (int off = 1; off < 16; off <<= 1)
    #pragma unroll
    for (int r = 0; r < 8; ++r) s[r] += __shfl_xor(s[r], off, 32);
  if ((lane & 15) == 0) {
    int rb = (lane >> 4) * 8;
    #pragma unroll
    for (int r = 0; r < 8; ++r) atomicAdd(&rowss[rb + r], s[r]);
  }
  __syncthreads();
  if (tid < 16) {
    float mxn = sqrtf(fmaxf(rowss[tid], EPSF));
    float u   = unS[tid];
    facS[tid] = tanhf(mxn / u * artanh_c(u)) / mxn;   // mobius_matvec scale
  }
  __syncthreads();

  const int n   = wave * 16 + (lane & 15);
  const int rhi = (lane >> 4) * 8;
  #pragma unroll
  for (int r = 0; r < 8; ++r) {
    int row = rhi + r;
    ux[(size_t)(rowbase + row) * G3 + gate * Hn + n] = d[r] * facS[row];
  }
}

// ---------------- Stage 3: sequential Mobius GRU scan ----------------
// grid = 4 blocks x 512 threads; block owns batch rows [bid*16, bid*16+16).
__global__ __launch_bounds__(512) void k_scan(
    const float* __restrict__ ux, const float* __restrict__ whh,
    const float* __restrict__ bias, const float* __restrict__ h0,
    float* __restrict__ out) {
  constexpr int LDH = Hn + 4;
  __shared__ float hS [16 * LDH];       // current hidden state
  __shared__ float rhS[16 * LDH];       // r_t (*) h  (phase-2 A operand)
  __shared__ float mxS[2][16 * LDH];    // raw matvec results
  __shared__ float biasS[G3];
  __shared__ float hnS[16];

  const int rowbase = blockIdx.x * 16;  // batch offset
  const int tid  = threadIdx.x;
  const int wave = tid >> 5, lane = tid & 31;

  for (int idx = tid; idx < 16 * Hn; idx += 512)
    hS[(idx >> 8) * LDH + (idx & (Hn - 1))] = h0[(size_t)rowbase * Hn + idx];
  for (int idx = tid; idx < G3; idx += 512) biasS[idx] = bias[idx];
  __syncthreads();

  for (int t = 0; t < Tn; ++t) {
    // ||h|| per row: wave w reduces row w
    {
      float ss = 0.f;
      #pragma unroll
      for (int i = 0; i < 8; ++i) { float v = hS[wave * LDH + lane + 32 * i]; ss += v * v; }
      ss = wave_allred(ss);
      if (lane == 0) hnS[wave] = sqrtf(fmaxf(ss, EPSF));
    }
    __syncthreads();

    // phase 1: raw matvecs for r (whh rows 0..255) and z (rows 512..767)
    #pragma unroll
    for (int gi = 0; gi < 2; ++gi) {
      v8f d = wmma_16x256(hS, LDH, whh, (gi == 0 ? 0 : 2 * Hn) + wave * 16, lane);
      const int n = wave * 16 + (lane & 15);
      const int rhi = (lane >> 4) * 8;
      #pragma unroll
      for (int r = 0; r < 8; ++r) mxS[gi][(rhi + r) * LDH + n] = d[r];
    }
    __syncthreads();

    // wave w completes row w: gates r and z, then rh = r (*) h
    float hrow[8], rt[8], zt[8], rh[8];
    const float hn  = hnS[wave];
    const float ath = artanh_c(hn);
    const float* uxp = ux + (size_t)(t * Bn + rowbase + wave) * G3;
    #pragma unroll
    for (int i = 0; i < 8; ++i) hrow[i] = hS[wave * LDH + lane + 32 * i];

    #pragma unroll
    for (int gi = 0; gi < 2; ++gi) {              // gi=0 -> r gate, gi=1 -> z gate
      float mx[8], a[8], u[8], br[8], s1[8], p[8];
      const int gofs = (gi == 0) ? 0 : 2 * Hn;    // ux/bias gate offset (r=0,z=512)
      #pragma unroll
      for (int i = 0; i < 8; ++i) {
        mx[i] = mxS[gi][wave * LDH + lane + 32 * i];
        u[i]  = uxp[gofs + lane + 32 * i];
        br[i] = biasS[gofs + lane + 32 * i];
      }
      float mxn = sqrtf(fmaxf(dot8(mx, mx), EPSF));
      float fm  = tanhf(mxn / hn * ath) / mxn;    // finish mobius_matvec(W_h*, h)
      #pragma unroll
      for (int i = 0; i < 8; ++i) a[i] = mx[i] * fm;
      mob_add(a, u, s1);                          // (Wh) (+) (Ux)
      mob_add(s1, br, p);                         // (+) bias
      float pn = sqrtf(fmaxf(dot8(p, p), EPSF));  // logmap0 + sigmoid
      float fl = artanh_c(pn) / pn;
      #pragma unroll
      for (int i = 0; i < 8; ++i) {
        float g = 1.f / (1.f + expf(-fl * p[i]));
        if (gi == 0) rt[i] = g; else zt[i] = g;
      }
    }
    mob_pmul(rt, hrow, hn, rh);                   // rh_t = r_t (*) h
    #pragma unroll
    for (int i = 0; i < 8; ++i) rhS[wave * LDH + lane + 32 * i] = rh[i];
    __syncthreads();

    // phase 2: raw matvec W_hh (rows 256..511) applied to rh
    {
      v8f d = wmma_16x256(rhS, LDH, whh, Hn + wave * 16, lane);
      const int n = wave * 16 + (lane & 15);
      const int rhi = (lane >> 4) * 8;
      #pragma unroll
      for (int r = 0; r < 8; ++r) mxS[0][(rhi + r) * LDH + n] = d[r];
    }
    __syncthreads();

    // wave w completes row w: h_tilde, delta, h_new
    {
      float mh[8], a[8], u[8], br[8], s1[8], htil[8], negh[8], delta[8], zd[8], hnew[8];
      #pragma unroll
      for (int i = 0; i < 8; ++i) {
        mh[i] = mxS[0][wave * LDH + lane + 32 * i];
        u[i]  = uxp[Hn + lane + 32 * i];           // h-gate ux
        br[i] = biasS[Hn + lane + 32 * i];         // b_h
      }
      float rhn  = sqrtf(fmaxf(dot8(rh, rh), EPSF));
      float athr = artanh_c(rhn);
      float mxn  = sqrtf(fmaxf(dot8(mh, mh), EPSF));
      float fm   = tanhf(mxn / rhn * athr) / mxn;
      #pragma unroll
      for (int i = 0; i < 8; ++i) a[i] = mh[i] * fm;
      mob_add(a, u, s1);
      mob_add(s1, br, htil);
      #pragma unroll
      for (int i = 0; i < 8; ++i) negh[i] = -hrow[i];
      mob_add(negh, htil, delta);                  // delta_h = (-h) (+) h_tilde
      float dn = sqrtf(fmaxf(dot8(delta, delta), EPSF));
      mob_pmul(zt, delta, dn, zd);                 // z_t (*) delta_h
      mob_add(hrow, zd, hnew);                     // h (+) ...
      float* orow = out + (size_t)t * Bn * Hn + (size_t)(rowbase + wave) * Hn;
      #pragma unroll
      for (int i = 0; i < 8; ++i) {
        orow[lane + 32 * i] = hnew[i];
        hS[wave * LDH + lane + 32 * i] = hnew[i];
      }
    }
    __syncthreads();
  }

  // h_last (== outs[T-1]) appended after outs
  float* hl = out + (size_t)Tn * Bn * Hn + (size_t)(rowbase + wave) * Hn;
  #pragma unroll
  for (int i = 0; i < 8; ++i) hl[lane + 32 * i] = hS[wave * LDH + lane + 32 * i];
}

extern "C" void kernel_launch(void* const* d_in, const int* in_sizes, int n_in,
                              void* d_out, int out_size, void* d_ws, size_t ws_size,
                              hipStream_t stream) {
  (void)in_sizes; (void)n_in; (void)out_size; (void)ws_size;
  const float* x    = (const float*)d_in[0];
  const float* h0   = (const float*)d_in[1];
  const float* wih  = (const float*)d_in[2];
  const float* whh  = (const float*)d_in[3];
  const float* bias = (const float*)d_in[4];
  float* out = (float*)d_out;

  float* f  = (float*)d_ws;          // NR floats
  float* un = f + NR;                // NR floats
  float* ux = un + NR;               // NR * 768 floats (~192 MB)

  k_rownorm<<<NR / 8, 256, 0, stream>>>(x, f, un);
  k_gemm_ih<<<dim3(NR / 16, 3), 512, 0, stream>>>(x, wih, f, un, ux);
  k_scan<<<4, 512, 0, stream>>>(ux, whh, bias, h0, out);
}